// Align_Net_26542897889960
// MI455X (gfx1250) — compile-verified
//
#include <hip/hip_runtime.h>
#include <math.h>

typedef __attribute__((ext_vector_type(2))) float v2f;
typedef __attribute__((ext_vector_type(8))) float v8f;

#define DG   8
#define K2   9
#define HLO  48
#define WLO  48
#define HHI  96
#define WHI  96
#define OMCH 216      // DG*3*K2
#define MPAD 224      // OMCH padded to 14*16
#define INC  64
#define OUTC 64
#define KTOT 576      // 64*9 == DG*8*9
#define NK4  144      // KTOT/4

// Workspace layout (floats):
//   s1    : 4*216*48*48 = 1,990,656   (low-res conv output)
//   Apack : 144*224*4   =   129,024   (w_om packed [ks][m][4], zero-padded)
//   Wpack : 144*64*4    =    36,864   (w_dc packed [ks][o][4])
#define S1_ELEMS    (4 * OMCH * HLO * WLO)
#define APACK_ELEMS (NK4 * MPAD * 4)
#define WPACK_ELEMS (NK4 * OUTC * 4)

// ---------------------------------------------------------------------------
// Kernel 0: pack both weight tensors into [k-group][M][4] so the WMMA A
// operand is a guard-free coalesced float2 load (lane m, pair 2*hiK).
// ---------------------------------------------------------------------------
__global__ void __launch_bounds__(256)
pack_weights(const float* __restrict__ Wom, const float* __restrict__ Wdc,
             float* __restrict__ Apack, float* __restrict__ Wpack)
{
    const int idx = blockIdx.x * 256 + threadIdx.x;
    if (idx < APACK_ELEMS) {
        const int j    = idx & 3;
        const int rest = idx >> 2;
        const int m    = rest % MPAD;
        const int ks   = rest / MPAD;
        Apack[idx] = (m < OMCH) ? Wom[(size_t)m * KTOT + ks * 4 + j] : 0.f;
    }
    if (idx < WPACK_ELEMS) {
        const int j    = idx & 3;
        const int rest = idx >> 2;
        const int o    = rest & 63;
        const int ks   = rest >> 6;
        Wpack[idx] = Wdc[(size_t)o * KTOT + ks * 4 + j];
    }
}

// ---------------------------------------------------------------------------
// Kernel 1: 3x3 conv (pad 1) as implicit GEMM, V_WMMA_F32_16X16X4_F32.
// 7 waves / block; the im2col halo (64ch x 3x18) is staged into LDS with
// CDNA5 async global->LDS copies (ASYNCcnt) -- zero-pad lanes use a plain
// ds_store.  Each wave owns 2 M-tiles; inner loop has no branches/divides.
// ---------------------------------------------------------------------------
__global__ void __launch_bounds__(224)
conv_om_wmma(const float* __restrict__ X, const float* __restrict__ Apack,
             const float* __restrict__ Bom, float* __restrict__ S1)
{
    __shared__ float lx[INC * 54];   // 64ch * 3row * 18col = 13.5 KB
    __shared__ int   ofs[KTOT];      // k -> (c*54 + ky*18 + kx)

    const int tid  = threadIdx.x;
    const int wid  = tid >> 5;       // 0..6
    const int lane = tid & 31;
    const int pix  = lane & 15;
    const int hiK  = lane >> 4;
    const int nt   = blockIdx.x;     // 144 pixel tiles
    const int b    = blockIdx.y;     // 4
    const int h    = nt / 3;
    const int w0   = (nt % 3) * 16;

    const float* xb = X + (size_t)b * INC * HLO * WLO;

    // Stage the halo: valid elements via async global->LDS DMA (per-lane
    // EXEC selects transfers), padding elements via direct LDS store of 0.
    for (int i = tid; i < INC * 54; i += 224) {
        const int c  = i / 54;
        const int r  = i - c * 54;
        const int ry = r / 18;
        const int cx = r - ry * 18;
        const int yy = h + ry - 1;
        const int xx = w0 + cx - 1;
        const unsigned lds_off = (unsigned)(size_t)(&lx[i]);   // flat LDS aperture: [31:0] = LDS offset
        if (yy >= 0 && yy < HLO && xx >= 0 && xx < WLO) {
            const float* gsrc = xb + (c * HLO + yy) * WLO + xx;
            asm volatile("global_load_async_to_lds_b32 %0, %1, off"
                         :: "v"(lds_off), "v"(gsrc) : "memory");
        } else {
            lx[i] = 0.f;
        }
    }
    for (int kk = tid; kk < KTOT; kk += 224) {
        const int c   = kk / 9;
        const int tap = kk - c * 9;
        const int ky  = tap / 3;
        ofs[kk] = c * 54 + ky * 18 + (tap - ky * 3);
    }
    asm volatile("s_wait_asynccnt 0x0" ::: "memory");
    __syncthreads();

    const int m0 = wid * 16 + pix;          // M-tile wid     (always < 216)
    const int m1 = (wid + 7) * 16 + pix;    // M-tile wid+7   (padded rows = 0)
    v8f acc0 = {}, acc1 = {};

#pragma unroll 4
    for (int ks = 0; ks < NK4; ++ks) {
        const int kk0 = 4 * ks + (hiK << 1);
        const int o0 = ofs[kk0], o1 = ofs[kk0 + 1];
        v2f bvec = {lx[o0 + pix], lx[o1 + pix]};
        v2f a0 = *reinterpret_cast<const v2f*>(Apack + ((size_t)ks * MPAD + m0) * 4 + (hiK << 1));
        v2f a1 = *reinterpret_cast<const v2f*>(Apack + ((size_t)ks * MPAD + m1) * 4 + (hiK << 1));
        acc0 = __builtin_amdgcn_wmma_f32_16x16x4_f32(false, a0, false, bvec,
                                                     (short)0, acc0, false, false);
        acc1 = __builtin_amdgcn_wmma_f32_16x16x4_f32(false, a1, false, bvec,
                                                     (short)0, acc1, false, false);
    }

#pragma unroll
    for (int r = 0; r < 8; ++r) {
        const int mo0 = wid * 16 + r + (hiK << 3);
        S1[(((size_t)b * OMCH + mo0) * HLO + h) * WLO + w0 + pix] = acc0[r] + Bom[mo0];
        const int mo1 = (wid + 7) * 16 + r + (hiK << 3);
        if (mo1 < OMCH)
            S1[(((size_t)b * OMCH + mo1) * HLO + h) * WLO + w0 + pix] = acc1[r] + Bom[mo1];
    }
}

// Exact reproduction of the reference's separable bilinear 48->96 resize.
__device__ __forceinline__ float sample_lowres(const float* __restrict__ p, int h, int w) {
    float sy = fmaxf(0.5f * (float)h - 0.25f, 0.0f);
    float sx = fmaxf(0.5f * (float)w - 0.25f, 0.0f);
    int y0 = (int)floorf(sy), x0 = (int)floorf(sx);
    int y1 = min(y0 + 1, HLO - 1), x1 = min(x0 + 1, WLO - 1);
    float ty = sy - (float)y0, tx = sx - (float)x0;
    float v00 = p[y0 * WLO + x0], v01 = p[y0 * WLO + x1];
    float v10 = p[y1 * WLO + x0], v11 = p[y1 * WLO + x1];
    return v00 * (1.f - ty) * (1.f - tx) + v01 * (1.f - ty) * tx
         + v10 * ty * (1.f - tx) + v11 * ty * tx;
}

// ---------------------------------------------------------------------------
// Kernel 2: fused resize + sigmoid + modulated deformable gather + GEMM.
// One wave per 16-pixel tile; LDS sample table (mask/validity premultiplied
// weights + clamped indices) + packed (chan|samp) k-decomposition table.
// Inner loop: 1 ds table read, 2x ds_load_b128 samples, 8 gathers, 4 WMMA.
// ---------------------------------------------------------------------------
struct Samp { float w[4]; int idx[4]; };   // 32 bytes

__global__ void __launch_bounds__(32)
deform_wmma(const float* __restrict__ Y, const float* __restrict__ S1,
            const float* __restrict__ Wpack, const float* __restrict__ Bdc,
            float* __restrict__ Out)
{
    __shared__ Samp smp[DG * K2][16];   // 36 KB
    __shared__ int  tab[KTOT];          // (chan<<16) | sample-slot

    const int lane = threadIdx.x;
    const int pix  = lane & 15;
    const int hiK  = lane >> 4;
    const int tile = blockIdx.x;        // 576 tiles (96/16 = 6 per row)
    const int b    = blockIdx.y;        // 4
    const int h    = tile / 6;
    const int w0   = (tile % 6) * 16;

    __builtin_prefetch(Wpack + lane * 16, 0, 0);

    const float* lob = S1 + (size_t)b * OMCH * HLO * WLO;

    for (int kk = lane; kk < KTOT; kk += 32) {
        const int g   = kk / 72;
        const int rem = kk - g * 72;
        const int c   = rem / 9;
        const int k2  = rem - c * 9;
        tab[kk] = ((g * 8 + c) << 16) | (g * 9 + k2);
    }

    for (int e = lane; e < DG * K2 * 16; e += 32) {
        const int s  = e >> 4;
        const int p  = e & 15;
        const int wp = w0 + p;
        const int g  = s / 9;
        const int k2 = s - g * 9;
        const float offy = sample_lowres(lob + (size_t)(g * 18 + k2 * 2 + 0) * (HLO * WLO), h, wp);
        const float offx = sample_lowres(lob + (size_t)(g * 18 + k2 * 2 + 1) * (HLO * WLO), h, wp);
        const float mraw = sample_lowres(lob + (size_t)(DG * 2 * K2 + g * 9 + k2) * (HLO * WLO), h, wp);
        const float mm = 1.f / (1.f + __expf(-mraw));
        const float py = (float)h  + (float)(k2 / 3 - 1) + offy;
        const float px = (float)wp + (float)(k2 % 3 - 1) + offx;
        const float y0f = floorf(py), x0f = floorf(px);
        const float ty = py - y0f, tx = px - x0f;
        const int y0 = (int)y0f, x0 = (int)x0f;
        const int y1 = y0 + 1,   x1 = x0 + 1;
        const float vy0 = (y0 >= 0 && y0 < HHI) ? 1.f : 0.f;
        const float vy1 = (y1 >= 0 && y1 < HHI) ? 1.f : 0.f;
        const float vx0 = (x0 >= 0 && x0 < WHI) ? 1.f : 0.f;
        const float vx1 = (x1 >= 0 && x1 < WHI) ? 1.f : 0.f;
        Samp sm;
        sm.w[0] = (1.f - ty) * (1.f - tx) * mm * vy0 * vx0;
        sm.w[1] = (1.f - ty) * tx         * mm * vy0 * vx1;
        sm.w[2] = ty         * (1.f - tx) * mm * vy1 * vx0;
        sm.w[3] = ty         * tx         * mm * vy1 * vx1;
        const int yc0 = min(max(y0, 0), HHI - 1), yc1 = min(max(y1, 0), HHI - 1);
        const int xc0 = min(max(x0, 0), WHI - 1), xc1 = min(max(x1, 0), WHI - 1);
        sm.idx[0] = yc0 * WHI + xc0;
        sm.idx[1] = yc0 * WHI + xc1;
        sm.idx[2] = yc1 * WHI + xc0;
        sm.idx[3] = yc1 * WHI + xc1;
        smp[s][p] = sm;
    }
    __syncthreads();

    const float* Yb = Y + (size_t)b * INC * (HHI * WHI);
    v8f acc[4] = {{}, {}, {}, {}};

#pragma unroll 2
    for (int ks = 0; ks < NK4; ++ks) {
        const int kk0 = 4 * ks + (hiK << 1);
        const int t0 = tab[kk0], t1 = tab[kk0 + 1];
        float bvs[2];
#pragma unroll
        for (int j = 0; j < 2; ++j) {
            const int t = j ? t1 : t0;
            const Samp sm = smp[t & 0xffff][pix];
            const float* yc = Yb + (size_t)(t >> 16) * (HHI * WHI);
            bvs[j] = sm.w[0] * yc[sm.idx[0]] + sm.w[1] * yc[sm.idx[1]]
                   + sm.w[2] * yc[sm.idx[2]] + sm.w[3] * yc[sm.idx[3]];
        }
        v2f bvec = {bvs[0], bvs[1]};
#pragma unroll
        for (int mt = 0; mt < 4; ++mt) {
            v2f a = *reinterpret_cast<const v2f*>(
                Wpack + ((size_t)ks * OUTC + mt * 16 + pix) * 4 + (hiK << 1));
            acc[mt] = __builtin_amdgcn_wmma_f32_16x16x4_f32(false, a, false, bvec,
                                                            (short)0, acc[mt], false, false);
        }
    }

#pragma unroll
    for (int mt = 0; mt < 4; ++mt) {
#pragma unroll
        for (int r = 0; r < 8; ++r) {
            const int o = mt * 16 + r + (hiK << 3);
            float v = acc[mt][r] + Bdc[o];
            v = fmaxf(v, 0.f);
            Out[(((size_t)b * OUTC + o) * HHI + h) * WHI + w0 + pix] = v;
        }
    }
}

extern "C" void kernel_launch(void* const* d_in, const int* in_sizes, int n_in,
                              void* d_out, int out_size, void* d_ws, size_t ws_size,
                              hipStream_t stream) {
    (void)in_sizes; (void)n_in; (void)out_size; (void)ws_size;
    const float* x    = (const float*)d_in[0];
    const float* y    = (const float*)d_in[1];
    const float* w_om = (const float*)d_in[2];
    const float* b_om = (const float*)d_in[3];
    const float* w_dc = (const float*)d_in[4];
    const float* b_dc = (const float*)d_in[5];
    float* out   = (float*)d_out;
    float* s1    = (float*)d_ws;
    float* Apack = s1 + S1_ELEMS;
    float* Wpack = Apack + APACK_ELEMS;

    pack_weights<<<(APACK_ELEMS + 255) / 256, 256, 0, stream>>>(w_om, w_dc, Apack, Wpack);
    conv_om_wmma<<<dim3(144, 4), 224, 0, stream>>>(x, Apack, b_om, s1);
    deform_wmma<<<dim3(576, 4), 32, 0, stream>>>(y, s1, Wpack, b_dc, out);
}